// TorchSTFT_58488864637518
// MI455X (gfx1250) — compile-verified
//
#include <hip/hip_runtime.h>
#include <math.h>

typedef float v2f __attribute__((ext_vector_type(2)));
typedef float v8f __attribute__((ext_vector_type(8)));

#define N_FFT   800
#define HOP     200
#define NFREQ   401
#define KPAD    416            // 401 padded to multiple of 16
#define B_SZ    16
#define L_IN    400000
#define OUT_LEN 400800         // N_FFT + HOP*(F-1)
#define F_FRAMES 2001
#define M_BLK   64             // frames per workgroup (4 x 16-row M-subtiles)
#define MSUB    4
#define FTILES  32             // ceil(2001/64)
#define SPAN    (63*HOP + N_FFT)   // 13400
#define NTHREADS 512           // 16 wave32s

#define JTILES  (KPAD / 16)    // 26 freq tiles
#define TTILES  (N_FFT / 16)   // 50 time tiles
#define G_FWD   (N_FFT / 4)    // 200 k-steps forward
#define G_INV   (KPAD / 4)     // 104 k-steps inverse

// workspace layout (float offsets); basis arrays stored pre-swizzled in the
// WMMA B-operand register layout: flat = ((g*TILES + tile)*32 + lane)*2 + c
#define WS_FWD_C 0
#define WS_FWD_S (WS_FWD_C + G_FWD*JTILES*64)
#define WS_INV_C (WS_FWD_S + G_FWD*JTILES*64)
#define WS_INV_S (WS_INV_C + G_INV*TTILES*64)
#define WS_Y     (WS_INV_S + G_INV*TTILES*64)

#define TWO_PI 6.283185307179586

// Build both bases directly in WMMA B-operand order so the GEMM loops issue a
// single coalesced global_load_b64 per operand per k-step.
//   b.x = B[kb+0][n], b.y = B[kb+1][n], kb = (lane>>4)*2, n = lane&15
__global__ void build_basis_kernel(float* __restrict__ ws) {
    int idx = blockIdx.x * blockDim.x + threadIdx.x;
    if (idx >= G_FWD * JTILES * 64) return;   // == G_INV*TTILES*64 == 332800

    const int c    = idx & 1;
    const int lane = (idx >> 1) & 31;
    const int rest = idx >> 6;

    // ---- forward basis [n][k]: cos / -sin, zero-padded k>=401 ----
    {
        const int jt = rest % JTILES;
        const int g  = rest / JTILES;                 // 0..199
        const int n  = g * 4 + ((lane >> 4) << 1) + c; // time sample 0..799
        const int k  = jt * 16 + (lane & 15);          // freq bin 0..415
        float fc = 0.f, fs = 0.f;
        if (k < NFREQ) {
            int r = (n * k) % N_FFT;
            float ang = (float)(TWO_PI * (double)r / (double)N_FFT);
            float s, cc; sincosf(ang, &s, &cc);
            fc = cc; fs = -s;
        }
        ws[WS_FWD_C + idx] = fc;
        ws[WS_FWD_S + idx] = fs;
    }

    // ---- inverse basis [k][t]: one-sided-weighted cos / -sin, /N ----
    {
        const int tt = rest % TTILES;
        const int g  = rest / TTILES;                 // 0..103
        const int k  = g * 4 + ((lane >> 4) << 1) + c; // freq bin 0..415
        const int t  = tt * 16 + (lane & 15);          // time sample 0..799
        float ic = 0.f, is = 0.f;
        if (k < NFREQ) {
            float wgt = (k == 0 || k == NFREQ - 1) ? 1.0f : 2.0f;
            float scale = wgt / (float)N_FFT;
            int r = (k * t) % N_FFT;
            float ang = (float)(TWO_PI * (double)r / (double)N_FFT);
            float s, cc; sincosf(ang, &s, &cc);
            ic = scale * cc; is = -scale * s;
        }
        ws[WS_INV_C + idx] = ic;
        ws[WS_INV_S + idx] = is;
    }
}

// One workgroup = one (batch, 64-frame tile). 16 wave32s, ~270 KB LDS.
// Phase 1: STFT via V_WMMA_F32_16X16X4_F32 (4 M-subtiles share each B fetch),
//          polar roundtrip -> LDS spectrum.
// Phase 2: inverse DFT via WMMA + windowed overlap-add with f32 global atomics.
__launch_bounds__(NTHREADS)
__global__ void stft_roundtrip_kernel(const float* __restrict__ x,
                                      const float* __restrict__ window,
                                      float* __restrict__ ws) {
    __shared__ float sFrame[SPAN];          // 13400 f32
    __shared__ float sWin[N_FFT];           // 800 f32
    __shared__ float sRe[M_BLK * KPAD];     // 26624 f32
    __shared__ float sIm[M_BLK * KPAD];     // 26624 f32

    const int b   = blockIdx.y;
    const int f0  = blockIdx.x * M_BLK;
    const int tid = threadIdx.x;
    const int lane = tid & 31;
    const int wv   = tid >> 5;               // wave id 0..15
    const int m    = lane & 15;              // A-matrix row within subtile
    const int nn   = lane & 15;              // B/C/D column within tile
    const int kb   = (lane >> 4) << 1;       // K sub-offset {0,2} per ISA f32 A layout
    const int mrow = (lane >> 4) << 3;       // C/D row offset {0,8}

    const v2f* fwdCp = (const v2f*)(ws + WS_FWD_C);
    const v2f* fwdSp = (const v2f*)(ws + WS_FWD_S);
    const v2f* invCp = (const v2f*)(ws + WS_INV_C);
    const v2f* invSp = (const v2f*)(ws + WS_INV_S);
    float* yacc = ws + WS_Y;

    // stage reflect-padded input span + window into LDS
    const long s0 = (long)f0 * HOP;
    for (int i = tid; i < SPAN; i += NTHREADS) {
        long u = s0 + i - (N_FFT / 2);
        if (u < 0) u = -u;
        else if (u >= L_IN) u = 2L * L_IN - 2 - u;
        sFrame[i] = x[(size_t)b * L_IN + (size_t)u];
    }
    for (int i = tid; i < N_FFT; i += NTHREADS) sWin[i] = window[i];
    __syncthreads();

    // ---- Phase 1: forward DFT (windowed frames x basis), K = 800 ----
    for (int jt = wv; jt < JTILES; jt += NTHREADS / 32) {
        v8f accRe[MSUB], accIm[MSUB];
        {
            v8f z = {};
            #pragma unroll
            for (int s = 0; s < MSUB; ++s) { accRe[s] = z; accIm[s] = z; }
        }
        const int kf = jt * 16 + nn;
        for (int g = 0; g < G_FWD; ++g) {
            const int ib = (g * JTILES + jt) * 32 + lane;  // coalesced b64 per wave
            v2f bc = fwdCp[ib];
            v2f bs = fwdSp[ib];
            // linear walk: prefetch 16 k-steps (one+ cacheline run) ahead
            __builtin_prefetch(&fwdCp[ib + 16 * JTILES * 32], 0, 0);
            __builtin_prefetch(&fwdSp[ib + 16 * JTILES * 32], 0, 0);
            const int ka = g * 4 + kb;
            const float w0 = sWin[ka], w1 = sWin[ka + 1];
            #pragma unroll
            for (int s = 0; s < MSUB; ++s) {
                v2f a;
                a.x = sFrame[(s * 16 + m) * HOP + ka]     * w0;
                a.y = sFrame[(s * 16 + m) * HOP + ka + 1] * w1;
                accRe[s] = __builtin_amdgcn_wmma_f32_16x16x4_f32(
                    false, a, false, bc, (short)0, accRe[s], false, false);
                accIm[s] = __builtin_amdgcn_wmma_f32_16x16x4_f32(
                    false, a, false, bs, (short)0, accIm[s], false, false);
            }
        }
        // magnitude/phase roundtrip (matches reference's abs/angle/exp path)
        #pragma unroll
        for (int s = 0; s < MSUB; ++s) {
            #pragma unroll
            for (int r = 0; r < 8; ++r) {
                float re = accRe[s][r], im = accIm[s][r];
                float mag = sqrtf(re * re + im * im);
                float ph  = atan2f(im, re);
                float sn, cs; sincosf(ph, &sn, &cs);
                int mm = s * 16 + r + mrow;
                sRe[mm * KPAD + kf] = mag * cs;
                sIm[mm * KPAD + kf] = mag * sn;
            }
        }
    }
    __syncthreads();

    // ---- Phase 2: inverse DFT, K = 416, + windowed overlap-add ----
    for (int tt = wv; tt < TTILES; tt += NTHREADS / 32) {
        v8f acc[MSUB];
        {
            v8f z = {};
            #pragma unroll
            for (int s = 0; s < MSUB; ++s) acc[s] = z;
        }
        const int t = tt * 16 + nn;
        for (int g = 0; g < G_INV; ++g) {
            const int ib = (g * TTILES + tt) * 32 + lane;
            v2f bC = invCp[ib];
            v2f bS = invSp[ib];
            __builtin_prefetch(&invCp[ib + 16 * TTILES * 32], 0, 0);
            __builtin_prefetch(&invSp[ib + 16 * TTILES * 32], 0, 0);
            const int ka = g * 4 + kb;
            #pragma unroll
            for (int s = 0; s < MSUB; ++s) {
                v2f aR, aI;
                aR.x = sRe[(s * 16 + m) * KPAD + ka];
                aR.y = sRe[(s * 16 + m) * KPAD + ka + 1];
                aI.x = sIm[(s * 16 + m) * KPAD + ka];
                aI.y = sIm[(s * 16 + m) * KPAD + ka + 1];
                acc[s] = __builtin_amdgcn_wmma_f32_16x16x4_f32(
                    false, aR, false, bC, (short)0, acc[s], false, false);
                acc[s] = __builtin_amdgcn_wmma_f32_16x16x4_f32(
                    false, aI, false, bS, (short)0, acc[s], false, false);
            }
        }
        const float wt = sWin[t];
        #pragma unroll
        for (int s = 0; s < MSUB; ++s) {
            #pragma unroll
            for (int r = 0; r < 8; ++r) {
                int f = f0 + s * 16 + r + mrow;
                if (f < F_FRAMES) {
                    unsafeAtomicAdd(&yacc[(size_t)b * OUT_LEN + (size_t)f * HOP + t],
                                    acc[s][r] * wt);
                }
            }
        }
    }
}

// Divide by summed squared window (<=4 overlapping frames), trim pads, write out.
__global__ void finalize_kernel(const float* __restrict__ yacc,
                                const float* __restrict__ window,
                                float* __restrict__ out) {
    int idx = blockIdx.x * blockDim.x + threadIdx.x;
    if (idx >= B_SZ * L_IN) return;
    int b  = idx / L_IN;
    int tp = idx % L_IN;
    int pos = tp + N_FFT / 2;

    int fhi = pos / HOP;              if (fhi > F_FRAMES - 1) fhi = F_FRAMES - 1;
    int flo = (pos - (N_FFT - 1) + (HOP - 1)) / HOP;  if (flo < 0) flo = 0;

    float s = 0.f;
    for (int f = flo; f <= fhi; ++f) {
        float w = window[pos - f * HOP];
        s += w * w;
    }
    float y = yacc[(size_t)b * OUT_LEN + pos];
    out[idx] = y / (s > 1e-11f ? s : 1.0f);
}

extern "C" void kernel_launch(void* const* d_in, const int* in_sizes, int n_in,
                              void* d_out, int out_size, void* d_ws, size_t ws_size,
                              hipStream_t stream) {
    const float* x      = (const float*)d_in[0];   // [16, 400000] f32
    const float* window = (const float*)d_in[1];   // [800] f32
    float* out = (float*)d_out;                    // [16, 1, 400000] f32
    float* ws  = (float*)d_ws;

    // zero the overlap-add accumulator every call (graph-capture safe)
    hipMemsetAsync(ws + WS_Y, 0, (size_t)B_SZ * OUT_LEN * sizeof(float), stream);

    build_basis_kernel<<<(G_FWD * JTILES * 64 + 255) / 256, 256, 0, stream>>>(ws);

    dim3 grid(FTILES, B_SZ);
    stft_roundtrip_kernel<<<grid, NTHREADS, 0, stream>>>(x, window, ws);

    finalize_kernel<<<((B_SZ * L_IN) + 255) / 256, 256, 0, stream>>>(
        ws + WS_Y, window, out);
}